// GATLayer_15710990368898
// MI455X (gfx1250) — compile-verified
//
#include <hip/hip_runtime.h>
#include <math.h>

// ---------------------------------------------------------------------------
// GATv2 layer for MI455X (gfx1250, wave32).
// N=50000 nodes, E=1.6M edges, 128 features, 4 heads x 32 channels.
// Node projections use V_WMMA_F32_16X16X4_F32 (exact fp32 matrix path).
// Edge passes are L2-resident gathers + f32 atomics (192MB L2 holds all
// intermediates), so HBM traffic is ~200MB -> ~10us floor @ 23.3 TB/s.
// ---------------------------------------------------------------------------

constexpr int       N_NODES = 50000;
constexpr long long N_EDGES = 1600000;
constexpr int       DIM     = 128;   // IN_DIM == OUT_DIM == HEADS*C
constexpr int       HEADS   = 4;
constexpr int       CHC     = 32;    // channels per head
constexpr int       EDIM    = 16;

typedef float v2f __attribute__((ext_vector_type(2)));
typedef float v8f __attribute__((ext_vector_type(8)));

// ---------------------------------------------------------------------------
// Kernel 1: Y = X @ W + b   (X: [N,128], W: [128,128] row-major, Y: [N,128])
// Block = 256 threads = 8 waves; wave w owns the 16x16 tile at cols w*16.
// Fragment layouts per CDNA5 ISA 7.12.2 (32-bit operands, wave32):
//   A 16x4 : v0 = A[M=lane%16][K=2*(lane/16)],   v1 = same row, K+1
//   B 4x16 : v0 = B[K=2*(lane/16)][N=lane%16],   v1 = row K+1, same col
//   C 16x16: vgpr i = C[M = i + 8*(lane/16)][N = lane%16]
// ---------------------------------------------------------------------------
__global__ void proj_wmma_kernel(const float* __restrict__ X,
                                 const float* __restrict__ W,
                                 const float* __restrict__ bias,
                                 float* __restrict__ Y) {
  const int wave  = threadIdx.x >> 5;       // 0..7 -> column tile
  const int lane  = threadIdx.x & 31;
  const int lhalf = lane >> 4;              // 0 or 1
  const int l16   = lane & 15;
  const int row0  = blockIdx.x * 16;        // N_NODES = 3125 * 16, exact
  const int col0  = wave * 16;

  const float* Arow = X + (size_t)(row0 + l16) * DIM + 2 * lhalf;
  const float* Bcol = W + (size_t)(2 * lhalf) * DIM + (col0 + l16);

  v8f acc = {};
#pragma unroll
  for (int k = 0; k < DIM; k += 4) {
    v2f a, b;
    a.x = Arow[k];                          // A[row][k + 2*lhalf + 0]
    a.y = Arow[k + 1];                      // A[row][k + 2*lhalf + 1]
    b.x = Bcol[(size_t)k * DIM];            // W[k + 2*lhalf + 0][col]
    b.y = Bcol[(size_t)(k + 1) * DIM];      // W[k + 2*lhalf + 1][col]
    acc = __builtin_amdgcn_wmma_f32_16x16x4_f32(
        /*neg_a=*/false, a, /*neg_b=*/false, b,
        /*c_mod=*/(short)0, acc, /*reuse_a=*/false, /*reuse_b=*/false);
  }

  const int   col = col0 + l16;
  const float bv  = bias[col];
  float* Yp = Y + (size_t)row0 * DIM + col;
#pragma unroll
  for (int i = 0; i < 8; ++i) {
    const int r = i + 8 * lhalf;
    Yp[(size_t)r * DIM] = acc[i] + bv;
  }
}

// ---------------------------------------------------------------------------
// Kernel 2: init m = -inf, denom = 0, out-accumulator (d_out) = 0
// ---------------------------------------------------------------------------
__global__ void init_kernel(float* __restrict__ mmax,
                            float* __restrict__ denom,
                            float* __restrict__ out) {
  const long long i = (long long)blockIdx.x * blockDim.x + threadIdx.x;
  if (i < (long long)N_NODES * DIM) out[i] = 0.0f;
  if (i < (long long)N_NODES * HEADS) {
    mmax[i]  = -__builtin_inff();
    denom[i] = 0.0f;
  }
}

// Sign-split float atomic max: correct for all finite values, init = -inf.
__device__ __forceinline__ void atomicMaxFloat(float* addr, float v) {
  if (v >= 0.0f) atomicMax((int*)addr, __float_as_int(v));
  else           atomicMin((unsigned int*)addr, __float_as_uint(v));
}

// ---------------------------------------------------------------------------
// Kernel 3: per (edge, head) attention logit + segment max.
//   z = x_l[src] + x_r[dst] + edge_attr @ W_e ; LeakyReLU(0.2) ; dot(att_h)
// W_e (8KB) and att (512B) staged in LDS.
// ---------------------------------------------------------------------------
__global__ void edge_logits_kernel(const long long* __restrict__ ei,
                                   const float* __restrict__ eattr,
                                   const float* __restrict__ xl,
                                   const float* __restrict__ xr,
                                   const float* __restrict__ We,
                                   const float* __restrict__ att,
                                   float* __restrict__ logits,
                                   float* __restrict__ mmax) {
  __shared__ float sWe[EDIM * DIM];
  __shared__ float sAtt[DIM];
  for (int i = threadIdx.x; i < EDIM * DIM; i += blockDim.x) sWe[i] = We[i];
  for (int i = threadIdx.x; i < DIM; i += blockDim.x) sAtt[i] = att[i];
  __syncthreads();

  const long long tid = (long long)blockIdx.x * blockDim.x + threadIdx.x;
  const long long e   = tid >> 2;
  const int       h   = (int)(tid & 3);
  if (e >= N_EDGES) return;

  const int src = (int)ei[e];
  const int dst = (int)ei[N_EDGES + e];

  float ea[EDIM];
#pragma unroll
  for (int j = 0; j < EDIM; ++j) ea[j] = eattr[e * EDIM + j];

  const float* xls = xl + (size_t)src * DIM + h * CHC;
  const float* xrd = xr + (size_t)dst * DIM + h * CHC;

  float acc = 0.0f;
#pragma unroll 4
  for (int c = 0; c < CHC; ++c) {
    float ef = 0.0f;
#pragma unroll
    for (int j = 0; j < EDIM; ++j)
      ef = fmaf(ea[j], sWe[j * DIM + h * CHC + c], ef);
    float z = xls[c] + xrd[c] + ef;
    z = (z >= 0.0f) ? z : 0.2f * z;              // LeakyReLU(0.2)
    acc = fmaf(z, sAtt[h * CHC + c], acc);
  }

  logits[e * HEADS + h] = acc;
  atomicMaxFloat(&mmax[(size_t)dst * HEADS + h], acc);
}

// ---------------------------------------------------------------------------
// Kernel 4: ex = exp(logit - m[dst]) stored in-place; denom[dst] += ex
// ---------------------------------------------------------------------------
__global__ void edge_expsum_kernel(const long long* __restrict__ ei,
                                   float* __restrict__ logits,
                                   const float* __restrict__ mmax,
                                   float* __restrict__ denom) {
  const long long tid = (long long)blockIdx.x * blockDim.x + threadIdx.x;
  const long long e   = tid >> 2;
  const int       h   = (int)(tid & 3);
  if (e >= N_EDGES) return;
  const int dst = (int)ei[N_EDGES + e];
  const float m = mmax[(size_t)dst * HEADS + h];    // finite: dst has >=1 edge
  const float ex = __expf(logits[e * HEADS + h] - m);
  logits[e * HEADS + h] = ex;
  atomicAdd(&denom[(size_t)dst * HEADS + h], ex);
}

// ---------------------------------------------------------------------------
// Kernel 5: one wave32 per edge; lane owns 4 consecutive channels (float4).
//   out[dst] += alpha[head(lane)] * x_l[src]   (f32 atomics, L2-resident)
// ---------------------------------------------------------------------------
__global__ void edge_agg_kernel(const long long* __restrict__ ei,
                                const float* __restrict__ ex,
                                const float* __restrict__ denom,
                                const float* __restrict__ xl,
                                float* __restrict__ out) {
  const long long gtid = (long long)blockIdx.x * blockDim.x + threadIdx.x;
  const long long e    = gtid >> 5;
  const int       lane = (int)(gtid & 31);
  if (e >= N_EDGES) return;
  const int src = (int)ei[e];
  const int dst = (int)ei[N_EDGES + e];
  const int h   = lane >> 3;                        // 4 cols -> same head
  const float alpha = ex[e * HEADS + h] /
                      (denom[(size_t)dst * HEADS + h] + 1e-16f);
  const float4 v = *(const float4*)(xl + (size_t)src * DIM + lane * 4);
  float* o = out + (size_t)dst * DIM + lane * 4;
  atomicAdd(o + 0, v.x * alpha);
  atomicAdd(o + 1, v.y * alpha);
  atomicAdd(o + 2, v.z * alpha);
  atomicAdd(o + 3, v.w * alpha);
}

// ---------------------------------------------------------------------------
// Kernel 6: wave32 per node; bias + LayerNorm (shfl_xor reduce) + ReLU + x
// ---------------------------------------------------------------------------
__global__ void finalize_kernel(const float* __restrict__ x,
                                const float* __restrict__ conv_bias,
                                const float* __restrict__ gamma,
                                const float* __restrict__ beta,
                                float* __restrict__ out) {
  const long long gtid = (long long)blockIdx.x * blockDim.x + threadIdx.x;
  const long long node = gtid >> 5;
  const int       lane = (int)(gtid & 31);
  if (node >= N_NODES) return;

  float4 v  = *(float4*)(out + (size_t)node * DIM + lane * 4);
  const float4 cb = *(const float4*)(conv_bias + lane * 4);
  v.x += cb.x; v.y += cb.y; v.z += cb.z; v.w += cb.w;

  float s  = v.x + v.y + v.z + v.w;
  float s2 = v.x * v.x + v.y * v.y + v.z * v.z + v.w * v.w;
#pragma unroll
  for (int off = 16; off >= 1; off >>= 1) {
    s  += __shfl_xor(s,  off, 32);
    s2 += __shfl_xor(s2, off, 32);
  }
  const float mu  = s * (1.0f / DIM);
  const float var = s2 * (1.0f / DIM) - mu * mu;
  const float inv = rsqrtf(var + 1e-5f);

  const float4 g  = *(const float4*)(gamma + lane * 4);
  const float4 b  = *(const float4*)(beta + lane * 4);
  const float4 xv = *(const float4*)(x + (size_t)node * DIM + lane * 4);

  float4 r;
  r.x = fmaxf((v.x - mu) * inv * g.x + b.x, 0.0f) + xv.x;
  r.y = fmaxf((v.y - mu) * inv * g.y + b.y, 0.0f) + xv.y;
  r.z = fmaxf((v.z - mu) * inv * g.z + b.z, 0.0f) + xv.z;
  r.w = fmaxf((v.w - mu) * inv * g.w + b.w, 0.0f) + xv.w;
  *(float4*)(out + (size_t)node * DIM + lane * 4) = r;
}

// ---------------------------------------------------------------------------
extern "C" void kernel_launch(void* const* d_in, const int* in_sizes, int n_in,
                              void* d_out, int out_size, void* d_ws, size_t ws_size,
                              hipStream_t stream) {
  const float*     x         = (const float*)d_in[0];
  const long long* ei        = (const long long*)d_in[1];   // int64 [2,E]
  const float*     eattr     = (const float*)d_in[2];
  const float*     W_l       = (const float*)d_in[3];
  const float*     b_l       = (const float*)d_in[4];
  const float*     W_r       = (const float*)d_in[5];
  const float*     b_r       = (const float*)d_in[6];
  const float*     W_e       = (const float*)d_in[7];
  const float*     att       = (const float*)d_in[8];
  const float*     conv_bias = (const float*)d_in[9];
  const float*     gamma     = (const float*)d_in[10];
  const float*     beta      = (const float*)d_in[11];
  float*           out       = (float*)d_out;

  // Workspace layout (floats): x_l | x_r | logits/ex | m | denom  (~78.5 MB)
  float* ws = (float*)d_ws;
  float* xl = ws;
  float* xr = xl + (size_t)N_NODES * DIM;
  float* lg = xr + (size_t)N_NODES * DIM;
  float* mx = lg + (size_t)N_EDGES * HEADS;
  float* dn = mx + (size_t)N_NODES * HEADS;

  const int TB = 256;

  // 1) Node projections via f32 WMMA (exact precision)
  proj_wmma_kernel<<<N_NODES / 16, TB, 0, stream>>>(x, W_l, b_l, xl);
  proj_wmma_kernel<<<N_NODES / 16, TB, 0, stream>>>(x, W_r, b_r, xr);

  // 2) Init segment-max / denom / output accumulator
  {
    const long long total = (long long)N_NODES * DIM;
    init_kernel<<<(int)((total + TB - 1) / TB), TB, 0, stream>>>(mx, dn, out);
  }

  // 3) Edge logits + segment max
  {
    const long long total = N_EDGES * HEADS;
    edge_logits_kernel<<<(int)((total + TB - 1) / TB), TB, 0, stream>>>(
        ei, eattr, xl, xr, W_e, att, lg, mx);
  }

  // 4) exp + segment sum
  {
    const long long total = N_EDGES * HEADS;
    edge_expsum_kernel<<<(int)((total + TB - 1) / TB), TB, 0, stream>>>(
        ei, lg, mx, dn);
  }

  // 5) Weighted aggregation (wave per edge)
  {
    const long long total = N_EDGES * 32;
    edge_agg_kernel<<<(int)((total + TB - 1) / TB), TB, 0, stream>>>(
        ei, lg, dn, xl, out);
  }

  // 6) Bias + LayerNorm + ReLU + residual (wave per node)
  {
    const long long total = (long long)N_NODES * 32;
    finalize_kernel<<<(int)((total + TB - 1) / TB), TB, 0, stream>>>(
        x, conv_bias, gamma, beta, out);
  }
}